// MUTAGModel_64312840290834
// MI455X (gfx1250) — compile-verified
//
#include <hip/hip_runtime.h>

typedef __attribute__((ext_vector_type(2))) float v2f;
typedef __attribute__((ext_vector_type(8))) float v8f;

#define HID 128

// ---------------- degree / normalization precompute ----------------

__global__ void deg_init_k(float* deg, int n) {
  int i = blockIdx.x * blockDim.x + threadIdx.x;
  if (i < n) deg[i] = 1.0f;  // self-loop
}

__global__ void deg_count_k(const int* __restrict__ dst, float* deg, int E) {
  int e = blockIdx.x * blockDim.x + threadIdx.x;
  if (e < E) atomicAdd(&deg[dst[e]], 1.0f);
}

__global__ void deg_finish_k(const float* __restrict__ deg, float* dis, float* selfnorm, int n) {
  int i = blockIdx.x * blockDim.x + threadIdx.x;
  if (i < n) {
    float d = __frsqrt_rn(deg[i]);
    dis[i] = d;
    selfnorm[i] = d * d;
  }
}

__global__ void edge_norm_k(const int* __restrict__ src, const int* __restrict__ dst,
                            const float* __restrict__ dis, float* __restrict__ nrm, int E) {
  int e = blockIdx.x * blockDim.x + threadIdx.x;
  if (e < E) nrm[e] = dis[src[e]] * dis[dst[e]];
}

// ---------------- zero-pad layer-1 inputs to K=8 ----------------

__global__ void pad_x_k(const float* __restrict__ x, float* __restrict__ xpad,
                        int K1, int total8) {
  int i = blockIdx.x * blockDim.x + threadIdx.x;
  if (i >= total8) return;
  int node = i >> 3, k = i & 7;
  xpad[i] = (k < K1) ? x[node * K1 + k] : 0.0f;
}

__global__ void pad_w_k(const float* __restrict__ W1, float* __restrict__ wpad, int K1) {
  int i = blockIdx.x * blockDim.x + threadIdx.x;  // 8*128 threads
  if (i >= 8 * HID) return;
  int k = i >> 7;
  wpad[i] = (k < K1) ? W1[i] : 0.0f;  // rows 0..K1-1 copied verbatim (row-major K x 128)
}

// ---------------- dense GEMM via f32 WMMA, compile-time K, no hot-path guards ----
// C[N x 128] = A[N x K] @ W[K x 128]; block = 16 rows, wave w = 16-col tile.
template <int K>
__global__ __launch_bounds__(256) void gcn_gemm_wmma(
    const float* __restrict__ A, const float* __restrict__ W,
    float* __restrict__ C, int N) {
  const int wave = threadIdx.x >> 5;
  const int lane = threadIdx.x & 31;
  const int row0 = blockIdx.x << 4;
  const int col0 = wave << 4;
  const int r  = lane & 15;
  const int kh = lane >> 4;  // which K-half of the fragment this lane group holds

  int rr = row0 + r;
  if (rr >= N) rr = N - 1;  // clamp loads only; EXEC stays all-ones
  const float* arow = A + (size_t)rr * K + (kh << 1);
  const float* wcol = W + (size_t)(kh << 1) * HID + col0 + r;

  v8f acc = {};
#pragma unroll
  for (int k0 = 0; k0 < K; k0 += 4) {
    v2f a = *(const v2f*)(arow + k0);          // contiguous b64 load
    v2f b;
    b.x = wcol[k0 * HID];
    b.y = wcol[k0 * HID + HID];
    acc = __builtin_amdgcn_wmma_f32_16x16x4_f32(false, a, false, b,
                                                (short)0, acc, false, false);
  }

  const int mbase = row0 + (kh << 3);
  float* crow = C + (size_t)mbase * HID + col0 + r;
  if (row0 + 16 <= N) {  // uniform branch: full tile (always true when N % 16 == 0)
#pragma unroll
    for (int v = 0; v < 8; ++v) crow[v * HID] = acc[v];
  } else {
#pragma unroll
    for (int v = 0; v < 8; ++v)
      if (mbase + v < N) crow[v * HID] = acc[v];
  }
}

// ---------------- self-loop seed: agg = hw * dis^2 ----------------
__global__ void agg_init_k(const float* __restrict__ hw, const float* __restrict__ selfnorm,
                           float* __restrict__ agg, int total) {
  int i = blockIdx.x * blockDim.x + threadIdx.x;
  if (i < total) agg[i] = hw[i] * selfnorm[i >> 7];
}

// ---------------- edge scatter: one wave per edge, float4 per lane ----------------
__global__ __launch_bounds__(256) void gcn_scatter(
    const float* __restrict__ hw, const int* __restrict__ src,
    const int* __restrict__ dst, const float* __restrict__ nrm,
    float* __restrict__ agg, int E) {
  const int e = blockIdx.x * 8 + (threadIdx.x >> 5);
  if (e >= E) return;
  const int lane = threadIdx.x & 31;
  const int s = src[e];
  const int d = dst[e];
  const float nv = nrm[e];
  const float4 val = ((const float4*)(hw + (size_t)s * HID))[lane];
  float* out = agg + (size_t)d * HID + lane * 4;
  atomicAdd(out + 0, val.x * nv);
  atomicAdd(out + 1, val.y * nv);
  atomicAdd(out + 2, val.z * nv);
  atomicAdd(out + 3, val.w * nv);
}

// ---------------- bias + relu in place ----------------
__global__ void relu_bias_k(float* __restrict__ h, const float* __restrict__ b, int total) {
  int i = blockIdx.x * blockDim.x + threadIdx.x;
  if (i < total) {
    float v = h[i] + b[i & (HID - 1)];
    h[i] = v > 0.0f ? v : 0.0f;
  }
}

// ---------------- pooled buffer zero ----------------
__global__ void pool_zero_k(unsigned* pooled, int total) {
  int i = blockIdx.x * blockDim.x + threadIdx.x;
  if (i < total) pooled[i] = 0u;  // 0.0f bits; valid identity since h >= 0 post-relu
}

// ---------------- segment max pool (uint-bit atomicMax, h >= 0) ----------------
__global__ void gcn_pool(const float* __restrict__ h, const int* __restrict__ batch,
                         unsigned* __restrict__ pooled, int total) {
  int i = blockIdx.x * blockDim.x + threadIdx.x;
  if (i >= total) return;
  int node = i >> 7, f = i & (HID - 1);
  atomicMax(&pooled[(size_t)batch[node] * HID + f], __float_as_uint(h[i]));
}

// ---------------- final tiny linear: [G,128] @ [128,2] + bl ----------------
__global__ void gcn_final(const unsigned* __restrict__ pooled,
                          const float* __restrict__ Wl, const float* __restrict__ bl,
                          float* __restrict__ out, int G) {
  int idx = blockIdx.x * blockDim.x + threadIdx.x;
  if (idx >= G * 2) return;
  int g = idx >> 1, c = idx & 1;
  const unsigned* p = pooled + (size_t)g * HID;
  float s = bl[c];
#pragma unroll 8
  for (int f = 0; f < HID; ++f) s += __uint_as_float(p[f]) * Wl[f * 2 + c];
  out[idx] = s;
}

extern "C" void kernel_launch(void* const* d_in, const int* in_sizes, int n_in,
                              void* d_out, int out_size, void* d_ws, size_t ws_size,
                              hipStream_t stream) {
  const float* x   = (const float*)d_in[0];
  const int*  ei   = (const int*)d_in[1];
  const int*  batch= (const int*)d_in[2];
  const float* W1  = (const float*)d_in[3];
  const float* b1  = (const float*)d_in[4];
  const float* W2  = (const float*)d_in[5];
  const float* b2  = (const float*)d_in[6];
  const float* W3  = (const float*)d_in[7];
  const float* b3  = (const float*)d_in[8];
  const float* Wl  = (const float*)d_in[9];
  const float* bl  = (const float*)d_in[10];

  const int N  = in_sizes[2];
  const int E  = in_sizes[1] / 2;
  const int K1 = in_sizes[0] / N;        // 7
  const int G  = out_size / 2;           // 512

  const int* srcIdx = ei;
  const int* dstIdx = ei + E;

  // workspace carving (256B aligned chunks)
  char* ws = (char*)d_ws;
  size_t off = 0;
  auto take = [&](size_t bytes) -> char* {
    char* p = ws + off;
    off += (bytes + 255) & ~(size_t)255;
    return p;
  };
  const size_t NB = (size_t)N * HID * sizeof(float);  // 51.2 MB
  float*    hw       = (float*)take(NB);
  float*    agg      = (float*)take(NB);              // also next layer's input
  float*    xpad     = (float*)take((size_t)N * 8 * 4);
  float*    w1pad    = (float*)take((size_t)8 * HID * 4);
  float*    deg      = (float*)take((size_t)N * 4);
  float*    dis      = (float*)take((size_t)N * 4);
  float*    selfnorm = (float*)take((size_t)N * 4);
  float*    nrm      = (float*)take((size_t)E * 4);
  unsigned* pooled   = (unsigned*)take((size_t)G * HID * 4);

  const int T = 256;
  const int totNH = N * HID;

  // normalization + padding precompute
  deg_init_k<<<(N + T - 1) / T, T, 0, stream>>>(deg, N);
  deg_count_k<<<(E + T - 1) / T, T, 0, stream>>>(dstIdx, deg, E);
  deg_finish_k<<<(N + T - 1) / T, T, 0, stream>>>(deg, dis, selfnorm, N);
  edge_norm_k<<<(E + T - 1) / T, T, 0, stream>>>(srcIdx, dstIdx, dis, nrm, E);
  pad_x_k<<<(N * 8 + T - 1) / T, T, 0, stream>>>(x, xpad, K1, N * 8);
  pad_w_k<<<(8 * HID + T - 1) / T, T, 0, stream>>>(W1, w1pad, K1);
  pool_zero_k<<<(G * HID + T - 1) / T, T, 0, stream>>>(pooled, G * HID);

  const int gemmGrid    = (N + 15) / 16;
  const int scatterGrid = (E + 7) / 8;
  const int nhGrid      = (totNH + T - 1) / T;

  // ---- layer 1 (input xpad, K=8) ----
  gcn_gemm_wmma<8><<<gemmGrid, T, 0, stream>>>(xpad, w1pad, hw, N);
  agg_init_k<<<nhGrid, T, 0, stream>>>(hw, selfnorm, agg, totNH);
  gcn_scatter<<<scatterGrid, T, 0, stream>>>(hw, srcIdx, dstIdx, nrm, agg, E);
  relu_bias_k<<<nhGrid, T, 0, stream>>>(agg, b1, totNH);

  // ---- layer 2 (input agg, K=128) ----
  gcn_gemm_wmma<HID><<<gemmGrid, T, 0, stream>>>(agg, W2, hw, N);
  agg_init_k<<<nhGrid, T, 0, stream>>>(hw, selfnorm, agg, totNH);
  gcn_scatter<<<scatterGrid, T, 0, stream>>>(hw, srcIdx, dstIdx, nrm, agg, E);
  relu_bias_k<<<nhGrid, T, 0, stream>>>(agg, b2, totNH);

  // ---- layer 3 ----
  gcn_gemm_wmma<HID><<<gemmGrid, T, 0, stream>>>(agg, W3, hw, N);
  agg_init_k<<<nhGrid, T, 0, stream>>>(hw, selfnorm, agg, totNH);
  gcn_scatter<<<scatterGrid, T, 0, stream>>>(hw, srcIdx, dstIdx, nrm, agg, E);
  relu_bias_k<<<nhGrid, T, 0, stream>>>(agg, b3, totNH);

  // ---- pool + final linear ----
  gcn_pool<<<nhGrid, T, 0, stream>>>(agg, batch, pooled, totNH);
  gcn_final<<<(G * 2 + T - 1) / T, T, 0, stream>>>(pooled, Wl, bl, (float*)d_out, G);
}